// PCENstack_44624710206056
// MI455X (gfx1250) — compile-verified
//
#include <hip/hip_runtime.h>
#include <math.h>

typedef __attribute__((ext_vector_type(2))) float v2f;
typedef __attribute__((ext_vector_type(4))) float v4f;
typedef __attribute__((ext_vector_type(8))) float v8f;

#define WAVES_PER_BLOCK 4
#define ROW_T 2048
#define NGROUP 8 /* ROW_T / 256 */

// One wave32 processes one (b,p,f) row of T=2048.
// Forward-backward EMA via V_WMMA_F32_16X16X4_F32:
//   within-chunk (16-step) causal scan  = L(16x16, lower-tri) x X(16x16 tile)
//   cross-chunk carries                 = weighted Hillis-Steele wave shuffle scan
// Row staging global->LDS uses GLOBAL_LOAD_ASYNC_TO_LDS_B128 (ASYNCcnt path).
__global__ __launch_bounds__(WAVES_PER_BLOCK * 32)
void pcen_stack_wmma_kernel(const float* __restrict__ x,
                            const float* __restrict__ i_sig_alpha,
                            const float* __restrict__ log_delta,
                            const float* __restrict__ i_sig_r,
                            const float* __restrict__ z_ks,
                            float* __restrict__ out)
{
    __shared__ __align__(16) float x_s[WAVES_PER_BLOCK][ROW_T]; // input row
    __shared__ __align__(16) float y_s[WAVES_PER_BLOCK][ROW_T]; // forward-pass result

    const int lane = threadIdx.x & 31;
    const int w    = threadIdx.x >> 5;
    const int rid  = blockIdx.x * WAVES_PER_BLOCK + w;

    const int f = rid & 127;
    const int p = (rid >> 7) & 3;

    // ---- stage row in LDS: async global->LDS, 16B per lane per issue ----
    {
        // low 32 bits of the generic pointer to __shared__ == LDS byte offset
        const unsigned ldsbase = (unsigned)(size_t)&x_s[w][0];
        const unsigned long long gbase =
            (unsigned long long)(const void*)(x + (size_t)rid * ROW_T);
        #pragma unroll
        for (int it = 0; it < ROW_T / 128; ++it) {
            const unsigned voff = (unsigned)((lane + 32 * it) * 16); // byte offset
            const unsigned ldsa = ldsbase + voff;
            const unsigned long long ga = gbase + voff;
            asm volatile("global_load_async_to_lds_b128 %0, %1, off"
                         :: "v"(ldsa), "v"(ga) : "memory");
        }
        asm volatile("s_wait_asynccnt 0x0" ::: "memory");
    }

    // ---- per-row scalar parameters (uniform across wave) ----
    const float alpha = __fdividef(1.f, 1.f + __expf(-i_sig_alpha[p]));
    const float delta = __expf(log_delta[p]);
    const float rr    = __fdividef(1.f, 1.f + __expf(-i_sig_r[p]));
    const float dpr   = __powf(delta, rr);

    float wk0, wk1, wk2, wk3;
    {
        const float z0 = z_ks[(p * 4 + 0) * 128 + f];
        const float z1 = z_ks[(p * 4 + 1) * 128 + f];
        const float z2 = z_ks[(p * 4 + 2) * 128 + f];
        const float z3 = z_ks[(p * 4 + 3) * 128 + f];
        const float zm = fmaxf(fmaxf(z0, z1), fmaxf(z2, z3));
        wk0 = __expf(z0 - zm); wk1 = __expf(z1 - zm);
        wk2 = __expf(z2 - zm); wk3 = __expf(z3 - zm);
        const float zs = __fdividef(1.f, wk0 + wk1 + wk2 + wk3);
        wk0 *= zs; wk1 *= zs; wk2 *= zs; wk3 *= zs;
    }

    // WMMA 32-bit operand striping (per ISA 7.12.2):
    //   A (16x4):  lanes 0-15 = rows 0-15 with K={0,1}; lanes 16-31 same rows, K={2,3}
    //   C/D:       VGPR r: lanes 0-15 -> row r, col=lane; lanes 16-31 -> row r+8
    //   B (4x16):  assumed mirror of A: VGPR0 = K{0|2} rows, VGPR1 = K{1|3} rows
    const int n   = lane & 15;              // tile column (chunk id) / A-matrix row
    const int sel = (lane >= 16) ? 2 : 0;   // K offset for this half-wave
    const int o   = (lane >= 16) ? 8 : 0;   // C/D row offset for this half-wave

    float acc[NGROUP][8];                   // weighted filtfilt, reversed-time layout
    #pragma unroll
    for (int g = 0; g < NGROUP; ++g)
        #pragma unroll
        for (int r = 0; r < 8; ++r) acc[g][r] = 0.f;

    #pragma unroll 1
    for (int k = 0; k < 4; ++k) {
        const float s   = (k == 0) ? 0.015f : (k == 1) ? 0.04f : (k == 2) ? 0.1f : 0.25f;
        const float wkk = (k == 0) ? wk0    : (k == 1) ? wk1   : (k == 2) ? wk2  : wk3;
        const float a   = 1.f - s;

        // A operand: L[i][j] = s * a^(i-j) for j<=i (causal 16-step scan operator)
        v2f A[4];
        #pragma unroll
        for (int gk = 0; gk < 4; ++gk) {
            const int c0 = 4 * gk + sel;
            A[gk].x = (c0     <= n) ? s * __powf(a, (float)(n - c0))     : 0.f;
            A[gk].y = (c0 + 1 <= n) ? s * __powf(a, (float)(n - c0 - 1)) : 0.f;
        }

        const float a16    = __powf(a, 16.f);
        const float a16_16 = __powf(a16, 16.f);
        const float a16d0 = a16, a16d1 = a16 * a16;
        const float a16d2 = a16d1 * a16d1, a16d3 = a16d2 * a16d2;
        const float a16_c = __powf(a16, (float)n);    // a16^(chunk id) for this lane
        float apw[8];                                  // a^(row+1) carry-injection weights
        apw[0] = __powf(a, (float)(o + 1));
        #pragma unroll
        for (int r = 1; r < 8; ++r) apw[r] = apw[r - 1] * a;

        // Cross-chunk carry scan + in-tile fixup. z[c] = local y[15] of chunk c.
        // yend[c] = a16^(c+1)*G + inclusive_decay_scan(z)[c]; carry_c = yend[c-1].
        auto scan_fixup = [&](v8f& C, float& G) {
            float z = __shfl(C[7], 16 + n, 32);        // broadcast row15 to both halves
            float u;
            u = __shfl_up(z, 1, 32); z += (n >= 1) ? a16d0 * u : 0.f;
            u = __shfl_up(z, 2, 32); z += (n >= 2) ? a16d1 * u : 0.f;
            u = __shfl_up(z, 4, 32); z += (n >= 4) ? a16d2 * u : 0.f;
            u = __shfl_up(z, 8, 32); z += (n >= 8) ? a16d3 * u : 0.f;
            const float zprev = __shfl_up(z, 1, 32);
            const float carry = (n == 0) ? G : fmaf(a16_c, G, zprev);
            #pragma unroll
            for (int r = 0; r < 8; ++r) C[r] = fmaf(apw[r], carry, C[r]);
            G = fmaf(a16_16, G, __shfl(z, 15, 32));    // carry out of this group
        };

        // ---------------- forward pass ----------------
        float G = x_s[w][0];                           // lfilter_zi: y[0] = x[0]
        #pragma unroll
        for (int g = 0; g < NGROUP; ++g) {
            v8f C = {0.f, 0.f, 0.f, 0.f, 0.f, 0.f, 0.f, 0.f};
            #pragma unroll
            for (int gk = 0; gk < 4; ++gk) {
                const v2f B = *(const v2f*)&x_s[w][256 * g + 16 * n + 4 * gk + sel];
                C = __builtin_amdgcn_wmma_f32_16x16x4_f32(false, A[gk], false, B,
                                                          (short)0, C, false, false);
            }
            scan_fixup(C, G);
            const int t0 = 256 * g + 16 * n + o;       // y[t], rows o..o+7 of chunk n
            *(v4f*)&y_s[w][t0]     = (v4f){C[0], C[1], C[2], C[3]};
            *(v4f*)&y_s[w][t0 + 4] = (v4f){C[4], C[5], C[6], C[7]};
        }

        // ---------------- backward pass (reversed yf) ----------------
        float Gb = y_s[w][ROW_T - 1];                  // yr[0] = yf[T-1]
        #pragma unroll
        for (int g = 0; g < NGROUP; ++g) {
            v8f C = {0.f, 0.f, 0.f, 0.f, 0.f, 0.f, 0.f, 0.f};
            #pragma unroll
            for (int gk = 0; gk < 4; ++gk) {
                const int rb = (ROW_T - 2) - 256 * g - 16 * n - 4 * gk - sel;
                const v2f t2 = *(const v2f*)&y_s[w][rb];  // descending pair -> swap
                v2f B; B.x = t2.y; B.y = t2.x;
                C = __builtin_amdgcn_wmma_f32_16x16x4_f32(false, A[gk], false, B,
                                                          (short)0, C, false, false);
            }
            scan_fixup(C, Gb);
            #pragma unroll
            for (int r = 0; r < 8; ++r) acc[g][r] = fmaf(wkk, C[r], acc[g][r]);
        }
    }

    // ---- PCEN nonlinearity + store (accumulator holds reversed time) ----
    // exp(-alpha*(log(eps) + log1p(M/eps))) == (M + eps)^(-alpha)   (exact)
    const float EPS = 1e-6f;
    float* og = out + (size_t)rid * ROW_T;
    #pragma unroll
    for (int g = 0; g < NGROUP; ++g) {
        const int tb = (ROW_T - 1) - 256 * g - 16 * n - o;  // t for r = 0
        float v[8];
        #pragma unroll
        for (int r = 0; r < 8; ++r) {
            const int t = tb - r;
            const float Mout = __powf(acc[g][r] + EPS, -alpha);
            const float xv = x_s[w][t];
            v[r] = __powf(fmaf(xv, Mout, delta), rr) - dpr;
        }
        // t descends with r; write two 16B chunks with reversed component order,
        // non-temporal (write-once stream, keep it out of L2's working set)
        __builtin_nontemporal_store((v4f){v[7], v[6], v[5], v[4]}, (v4f*)&og[tb - 7]);
        __builtin_nontemporal_store((v4f){v[3], v[2], v[1], v[0]}, (v4f*)&og[tb - 3]);
    }
}

extern "C" void kernel_launch(void* const* d_in, const int* in_sizes, int n_in,
                              void* d_out, int out_size, void* d_ws, size_t ws_size,
                              hipStream_t stream) {
    (void)n_in; (void)out_size; (void)d_ws; (void)ws_size;
    const float* x           = (const float*)d_in[0];
    const float* i_sig_alpha = (const float*)d_in[1];
    const float* log_delta   = (const float*)d_in[2];
    const float* i_sig_r     = (const float*)d_in[3];
    const float* z_ks        = (const float*)d_in[4];
    float* out = (float*)d_out;

    const int rows   = in_sizes[0] / ROW_T;            // B*P*F = 8192
    const int blocks = rows / WAVES_PER_BLOCK;         // 2048
    pcen_stack_wmma_kernel<<<dim3(blocks), dim3(WAVES_PER_BLOCK * 32), 0, stream>>>(
        x, i_sig_alpha, log_delta, i_sig_r, z_ks, out);
}